// OrdinalCrossEntropyLoss_61890478736096
// MI455X (gfx1250) — compile-verified
//
#include <hip/hip_runtime.h>

// OrdinalCrossEntropyLoss for MI455X (gfx1250, wave32).
// Math reduces to: out = sum(-log(pred[b,s,tgt]+1e-8)*mask) / (sum(mask)+1e-7).
// Memory-bound (~36 MiB @ 23.3 TB/s ~= 1.6 us). WMMA f32 16x16x4 with a
// ones-matrix is used as the wave-level reduction tree (D[m][n] = sum_k A[m][k]).

typedef float v2f __attribute__((ext_vector_type(2)));
typedef float v8f __attribute__((ext_vector_type(8)));

// Reduce one float per lane across the wave32. Returns the full-wave sum in
// every lane. Requires EXEC == all ones (callers guarantee full waves).
//
// A-matrix layout (f32 16x4, 2 VGPRs): lanes 0-15 hold row M=lane, K=0 (.x)
// and K=1 (.y); lanes 16-31 hold row M=lane-16, K=2 (.x) and K=3 (.y).
// With A.y = 0 and B = ones(4x16):  D[m][n] = acc[m] + acc[m+16] for all n.
// D layout: VGPR r, lanes 0-15 -> D[r][lane]; lanes 16-31 -> D[8+r][lane-16].
// Summing the 8 D components per lane gives the half-wave sums; one xor-16
// shuffle produces the full-wave sum.
__device__ __forceinline__ float wave_reduce_wmma(float v) {
    v2f a; a.x = v;    a.y = 0.0f;
    v2f b; b.x = 1.0f; b.y = 1.0f;
    v8f c = {};
    // 8 args: (neg_a, A, neg_b, B, c_mod, C, reuse_a, reuse_b)
    c = __builtin_amdgcn_wmma_f32_16x16x4_f32(
        /*neg_a=*/false, a, /*neg_b=*/false, b,
        /*c_mod=*/(short)0, c, /*reuse_a=*/false, /*reuse_b=*/false);
    float s = ((c[0] + c[1]) + (c[2] + c[3])) + ((c[4] + c[5]) + (c[6] + c[7]));
    s += __shfl_xor(s, 16, 32);   // combine the two half-wave sums
    return s;
}

__global__ __launch_bounds__(256)
void ocel_reduce_kernel(const float* __restrict__ pred,   // (B*S, 16)
                        const int*   __restrict__ tgt,    // (B*S)
                        const float* __restrict__ mask,   // (B*S)
                        float*       __restrict__ ws,     // (gridDim.x, 2)
                        int n) {
    const int tid    = blockIdx.x * blockDim.x + threadIdx.x;
    const int stride = gridDim.x * blockDim.x;

    float accL = 0.0f;   // sum of masked -log(p+eps)
    float accM = 0.0f;   // sum of mask

    for (int i = tid; i < n; i += stride) {
        const int   t = tgt[i];                       // coalesced 4B
        const float m = mask[i];                      // coalesced 4B
        const float p = pred[(size_t)i * 16 + t];     // gather, 1 of 16 floats
        accL += (-__logf(p + 1e-8f)) * m;             // v_log_f32 path
        accM += m;
    }

    // Wave-level reduction via v_wmma_f32_16x16x4_f32 (EXEC all ones: the
    // grid-stride loop exits uniformly; every launched thread reaches here).
    const float wl = wave_reduce_wmma(accL);
    const float wm = wave_reduce_wmma(accM);

    // Block-level combine in a fixed order (deterministic, no atomics).
    __shared__ float sL[8];
    __shared__ float sM[8];
    const int lane = threadIdx.x & 31;
    const int wave = threadIdx.x >> 5;
    if (lane == 0) { sL[wave] = wl; sM[wave] = wm; }
    __syncthreads();
    if (threadIdx.x == 0) {
        float bl = 0.0f, bm = 0.0f;
        #pragma unroll
        for (int w = 0; w < 8; ++w) { bl += sL[w]; bm += sM[w]; }
        ws[2 * (size_t)blockIdx.x + 0] = bl;
        ws[2 * (size_t)blockIdx.x + 1] = bm;
    }
}

__global__ __launch_bounds__(32)
void ocel_final_kernel(const float* __restrict__ ws, float* __restrict__ out,
                       int nblocks) {
    float accL = 0.0f, accM = 0.0f;
    // nblocks is a multiple of 32 -> uniform trip count, EXEC stays all ones.
    for (int b = threadIdx.x; b < nblocks; b += 32) {
        accL += ws[2 * (size_t)b + 0];
        accM += ws[2 * (size_t)b + 1];
    }
    const float tl = wave_reduce_wmma(accL);
    const float tm = wave_reduce_wmma(accM);
    if (threadIdx.x == 0) {
        out[0] = tl / (tm + 1e-7f);
    }
}

extern "C" void kernel_launch(void* const* d_in, const int* in_sizes, int n_in,
                              void* d_out, int out_size, void* d_ws, size_t ws_size,
                              hipStream_t stream) {
    const float* pred = (const float*)d_in[0];  // (B,S,16) f32
    const int*   tgt  = (const int*)d_in[1];    // (B,S)    int32 (JAX demotes i64)
    const float* mask = (const float*)d_in[2];  // (B,S)    f32
    float*       out  = (float*)d_out;          // scalar f32
    float*       ws   = (float*)d_ws;           // 256 * 2 floats = 2 KiB scratch

    const int n = in_sizes[1];                  // B*S = 524288
    const int threads = 256;
    const int blocks  = 256;                    // 65536 threads, 8 elems each

    ocel_reduce_kernel<<<blocks, threads, 0, stream>>>(pred, tgt, mask, ws, n);
    ocel_final_kernel<<<1, 32, 0, stream>>>(ws, out, blocks);
}